// BboxLoss_15187004359202
// MI455X (gfx1250) — compile-verified
//
#include <hip/hip_runtime.h>
#include <stdint.h>

#define B_ 32
#define P_ 1024
#define T_ 512

typedef float v2f __attribute__((ext_vector_type(2)));
typedef float v8f __attribute__((ext_vector_type(8)));
typedef int   v4i __attribute__((ext_vector_type(4)));

#define AS1 __attribute__((address_space(1)))
#define AS3 __attribute__((address_space(3)))

#ifndef __has_builtin
#define __has_builtin(x) 0
#endif
#if __has_builtin(__builtin_amdgcn_global_load_async_to_lds_b128)
#define USE_ASYNC_LDS 1
#endif

__device__ __forceinline__ float one_minus_iou(float4 t, float4 p) {
    float ix1 = fmaxf(t.x, p.x);
    float iy1 = fmaxf(t.y, p.y);
    float ix2 = fminf(t.z, p.z);
    float iy2 = fminf(t.w, p.w);
    float iw = fmaxf(ix2 - ix1, 0.0f);
    float ih = fmaxf(iy2 - iy1, 0.0f);
    float inter = iw * ih;
    float at = (t.z - t.x) * (t.w - t.y);
    float ap = (p.z - p.x) * (p.w - p.y);
    float iou = inter / (at + ap - inter + 1e-7f);
    return 1.0f - iou;
}

__device__ __forceinline__ float mask_of(float4 t) {
    float mx = fmaxf(fmaxf(t.x, t.y), fmaxf(t.z, t.w));
    return (mx != 0.0f) ? 1.0f : 0.0f;
}

// Phase 1: L[t, p0:p0+16] for 8 t's per block.  Each wave owns one t and a
// 16-wide p strip; the masked batch reduction (K=B=32) is 8 chained
// V_WMMA_F32_16X16X4_F32 ops (K=4 batches per step).
__global__ __launch_bounds__(256) void bbox_pair_loss_kernel(
    const float* __restrict__ pred, const float* __restrict__ tgt,
    float* __restrict__ Lmat)
{
    __shared__ float4 sp4[B_ * 16]; // pred boxes [b][j], j = p-in-strip
    __shared__ float4 st4[B_ * 8];  // tgt  boxes [b][w], w = t-in-tile

    const int tid = threadIdx.x;
    const int p0 = (blockIdx.x & 63) << 4;
    const int t0 = (blockIdx.x >> 6) << 3;
    const float4* predG = (const float4*)pred; // [b*P_ + p]
    const float4* tgtG  = (const float4*)tgt;  // [b*T_ + t]

    // ---- stage tiles into LDS ----
    {
        const int q0 = tid, q1 = tid + 256;          // 512 pred float4 chunks
        const int b0 = q0 >> 4, j0 = q0 & 15;
        const int b1 = q1 >> 4, j1 = q1 & 15;
        const int bt = tid >> 3, wt = tid & 7;       // 256 tgt float4 chunks
#ifdef USE_ASYNC_LDS
        __builtin_amdgcn_global_load_async_to_lds_b128(
            (AS1 v4i*)&predG[b0 * P_ + p0 + j0], (AS3 v4i*)&sp4[q0], 0, 0);
        __builtin_amdgcn_global_load_async_to_lds_b128(
            (AS1 v4i*)&predG[b1 * P_ + p0 + j1], (AS3 v4i*)&sp4[q1], 0, 0);
        __builtin_amdgcn_global_load_async_to_lds_b128(
            (AS1 v4i*)&tgtG[bt * T_ + t0 + wt],  (AS3 v4i*)&st4[tid], 0, 0);
#if __has_builtin(__builtin_amdgcn_s_wait_asynccnt)
        __builtin_amdgcn_s_wait_asynccnt(0);
#else
        asm volatile("s_wait_asynccnt 0" ::: "memory");
#endif
#else
        sp4[q0]  = predG[b0 * P_ + p0 + j0];
        sp4[q1]  = predG[b1 * P_ + p0 + j1];
        st4[tid] = tgtG[bt * T_ + t0 + wt];
#endif
    }
    __syncthreads();

    const int lane = tid & 31;
    const int w    = tid >> 5;      // wave index -> t offset in tile
    const int m    = lane & 15;     // A-matrix row (p within strip)
    const int kh   = lane >> 4;     // half-wave: K in {0,1} vs {2,3}
    const int tcur = t0 + w;

    // valid-target count for this t (lane == b), butterfly sum over wave
    float cnt = mask_of(st4[lane * 8 + w]);
#pragma unroll
    for (int d = 16; d > 0; d >>= 1)
        cnt += __shfl_xor(cnt, d, 32);
    const float inv = 1.0f / fmaxf(cnt, 1.0f);

    v8f acc = {0.f, 0.f, 0.f, 0.f, 0.f, 0.f, 0.f, 0.f};
#pragma unroll
    for (int c = 0; c < 8; ++c) {
        const int b0 = 4 * c + 2 * kh;   // K slot of A/B vgpr0 for this half-wave
        const int b1 = b0 + 1;           // K slot of vgpr1
        const float4 tb0 = st4[b0 * 8 + w];   // uniform per half-wave (broadcast)
        const float4 tb1 = st4[b1 * 8 + w];
        const float4 pb0 = sp4[b0 * 16 + m];
        const float4 pb1 = sp4[b1 * 16 + m];
        v2f a, bm;
        a.x  = one_minus_iou(tb0, pb0);
        a.y  = one_minus_iou(tb1, pb1);
        bm.x = mask_of(tb0);             // B[k,n] = mask[b_k] broadcast over N
        bm.y = mask_of(tb1);
        acc = __builtin_amdgcn_wmma_f32_16x16x4_f32(
            false, a, false, bm, (short)0, acc, false, false);
    }

    // D: vgpr j = (M=j, N=lane) lanes 0-15 / (M=j+8, N=lane-16) lanes 16-31.
    // Every column N is identical; lanes 0 and 16 hold all 16 row values.
    if ((lane & 15) == 0) {
        const int pbase = p0 + kh * 8;
        float* dst = Lmat + (size_t)tcur * P_ + pbase;
        float4 o0 = make_float4(acc[0] * inv, acc[1] * inv, acc[2] * inv, acc[3] * inv);
        float4 o1 = make_float4(acc[4] * inv, acc[5] * inv, acc[6] * inv, acc[7] * inv);
        ((float4*)dst)[0] = o0;
        ((float4*)dst)[1] = o1;
    }
}

// Phase 2: sequential greedy matching (single workgroup, 32 wave32s).
__global__ __launch_bounds__(1024) void greedy_match_kernel(
    const float* __restrict__ Lmat, float* __restrict__ out)
{
    const int p    = threadIdx.x;     // column 0..1023
    const int lane = p & 31;
    const int wv   = p >> 5;

    __shared__ float swv[32];
    __shared__ int   swi[32];
    __shared__ float matched[T_];
    __shared__ int   winner;

    bool avail = true;

    for (int t = 0; t < T_; ++t) {
        float v = avail ? Lmat[t * P_ + p] : __builtin_inff();
        int idx = p;
#pragma unroll
        for (int d = 16; d > 0; d >>= 1) {   // argmin, ties -> lowest index
            float ov = __shfl_xor(v, d, 32);
            int   oi = __shfl_xor(idx, d, 32);
            if (ov < v || (ov == v && oi < idx)) { v = ov; idx = oi; }
        }
        if (lane == 0) { swv[wv] = v; swi[wv] = idx; }
        __syncthreads();
        if (wv == 0) {
            float v2 = swv[lane];
            int   i2 = swi[lane];
#pragma unroll
            for (int d = 16; d > 0; d >>= 1) {
                float ov = __shfl_xor(v2, d, 32);
                int   oi = __shfl_xor(i2, d, 32);
                if (ov < v2 || (ov == v2 && oi < i2)) { v2 = ov; i2 = oi; }
            }
            if (lane == 0) { winner = i2; matched[t] = v2; }
        }
        __syncthreads();
        if (p == winner) avail = false;
        // next write to winner/swv happens after the next barrier -> no race
    }
    __syncthreads();

    // Final sum of 512 matched values via WMMA row-sums: D = A x ones + C.
    if (wv == 0) {
        const int m  = lane & 15;
        const int kh = lane >> 4;
        v8f acc = {0.f, 0.f, 0.f, 0.f, 0.f, 0.f, 0.f, 0.f};
        v2f ones; ones.x = 1.0f; ones.y = 1.0f;
#pragma unroll
        for (int c = 0; c < 8; ++c) {
            const int base = c * 64;
            v2f a;
            a.x = matched[base + m * 4 + 2 * kh + 0];
            a.y = matched[base + m * 4 + 2 * kh + 1];
            acc = __builtin_amdgcn_wmma_f32_16x16x4_f32(
                false, a, false, ones, (short)0, acc, false, false);
        }
        // lane 0 holds row sums M=0..7 in acc[0..7], lane 16 holds M=8..15
        float s = acc[0] + acc[1] + acc[2] + acc[3] + acc[4] + acc[5] + acc[6] + acc[7];
        s += __shfl_xor(s, 16, 32);
        if (lane == 0)
            out[0] = (s + (float)(P_ - T_)) / (float)P_;
    }
}

extern "C" void kernel_launch(void* const* d_in, const int* in_sizes, int n_in,
                              void* d_out, int out_size, void* d_ws, size_t ws_size,
                              hipStream_t stream) {
    (void)in_sizes; (void)n_in; (void)out_size; (void)ws_size;
    const float* pred = (const float*)d_in[0];   // (32, 1024, 4) f32
    const float* tgt  = (const float*)d_in[1];   // (32, 512, 4)  f32
    float* Lmat = (float*)d_ws;                  // T_*P_ floats = 2 MB scratch

    bbox_pair_loss_kernel<<<dim3((T_ / 8) * (P_ / 16)), dim3(256), 0, stream>>>(
        pred, tgt, Lmat);
    greedy_match_kernel<<<dim3(1), dim3(1024), 0, stream>>>(
        Lmat, (float*)d_out);
}